// StackedGCNDBLP_3307124818593
// MI455X (gfx1250) — compile-verified
//
#include <hip/hip_runtime.h>
#include <hip/hip_bf16.h>

#define AUTHOR_N 500000
#define PAPER_N  1000000
#define CONF_N   5000
#define UD 8
#define PD 4
#define CD 4
#define IC 8
#define HDIM 16

typedef __attribute__((ext_vector_type(2))) float v2f;
typedef __attribute__((ext_vector_type(8))) float v8f;

__device__ __forceinline__ float relu_f(float v) { return fmaxf(v, 0.0f); }

// ---------------- degree ----------------
__global__ void k_init_deg(float* __restrict__ deg, int n) {
    int i = blockIdx.x * blockDim.x + threadIdx.x;
    if (i < n) deg[i] = 1.0f;  // self loop
}

__global__ void k_count_deg(const int* __restrict__ col, float* __restrict__ deg, int E) {
    int e = blockIdx.x * blockDim.x + threadIdx.x;
    if (e < E) atomicAdd(&deg[col[e]], 1.0f);
}

// ---------------- per-node typed embedding -> x (N x 8), dis ----------------
__global__ void k_embed(const float* __restrict__ emb_a, const float* __restrict__ emb_k,
                        const float* __restrict__ W_u, const float* __restrict__ b_u,
                        const float* __restrict__ emb_p, const float* __restrict__ W_p,
                        const float* __restrict__ b_p, const float* __restrict__ emb_c,
                        const float* __restrict__ W_c, const float* __restrict__ b_c,
                        const int* __restrict__ feat, const float* __restrict__ deg,
                        float* __restrict__ dis, float* __restrict__ x, int n) {
    int i = blockIdx.x * blockDim.x + threadIdx.x;
    if (i >= n) return;
    int idx   = feat[3 * i + 0];
    int known = feat[3 * i + 1];
    int t     = feat[3 * i + 2];
    float xi[IC];
    if (t == 0) {
        int a = min(max(idx, 0), AUTHOR_N - 1);
        int k = min(max(known, 0), 1);
        float e[UD];
        #pragma unroll
        for (int d = 0; d < UD; ++d) e[d] = relu_f(emb_a[a * UD + d] + emb_k[k * UD + d]);
        #pragma unroll
        for (int c = 0; c < IC; ++c) {
            float s = b_u[c];
            #pragma unroll
            for (int d = 0; d < UD; ++d) s += e[d] * W_u[d * IC + c];
            xi[c] = s;
        }
    } else if (t == 1) {
        int a = min(max(idx, 0), PAPER_N - 1);
        float e[PD];
        #pragma unroll
        for (int d = 0; d < PD; ++d) e[d] = relu_f(emb_p[a * PD + d]);
        #pragma unroll
        for (int c = 0; c < IC; ++c) {
            float s = b_p[c];
            #pragma unroll
            for (int d = 0; d < PD; ++d) s += e[d] * W_p[d * IC + c];
            xi[c] = s;
        }
    } else {
        int a = min(max(idx, 0), CONF_N - 1);
        float e[CD];
        #pragma unroll
        for (int d = 0; d < CD; ++d) e[d] = relu_f(emb_c[a * CD + d]);
        #pragma unroll
        for (int c = 0; c < IC; ++c) {
            float s = b_c[c];
            #pragma unroll
            for (int d = 0; d < CD; ++d) s += e[d] * W_c[d * IC + c];
            xi[c] = s;
        }
    }
    #pragma unroll
    for (int c = 0; c < IC; ++c) x[i * IC + c] = xi[c];
    float dg = deg[i];
    dis[i] = (dg > 0.0f) ? rsqrtf(dg) : 0.0f;
}

// ---------------- y = dis * (x @ W0), via V_WMMA_F32_16X16X4_F32 ----------------
// One wave handles 16 nodes. K=8 done as two 16x16x4 WMMAs.
// A (16x4 f32): lane m in [0,16): M=m holds K={0,1}; lanes 16-31: K={2,3}.
// B (4x16 f32): VGPR j, lane l: K = 2*(l>>4) + j, N = l%16.
// D (16x16 f32): VGPR j, lane l: M = j + 8*(l>>4), N = l%16.
__global__ void k_xw0_wmma(const float* __restrict__ x, const float* __restrict__ dis,
                           const float* __restrict__ W0, float* __restrict__ y,
                           float* __restrict__ acc1, int n) {
    int lane = threadIdx.x & 31;
    int wave = blockIdx.x * (blockDim.x >> 5) + (threadIdx.x >> 5);
    int base = wave * 16;
    if (base >= n) return;  // wave-uniform: EXEC all 1s inside
    int m  = lane & 15;
    int hi = lane >> 4;
    int node = base + m;
    if (node >= n) node = n - 1;  // clamp loads; stores guarded below
    float dsc = dis[node];
    v8f acc = {};
    #pragma unroll
    for (int kc = 0; kc < 2; ++kc) {
        int k0 = kc * 4 + 2 * hi;
        v2f a, b;
        a.x = dsc * x[node * IC + k0];
        a.y = dsc * x[node * IC + k0 + 1];
        b.x = W0[(k0)     * HDIM + m];
        b.y = W0[(k0 + 1) * HDIM + m];
        acc = __builtin_amdgcn_wmma_f32_16x16x4_f32(false, a, false, b, (short)0, acc,
                                                    false, false);
    }
    int row0 = base + 8 * hi;
    if (base + 16 <= n) {
        // full tile: straight-line coalesced stores (16 cols -> 16 consecutive lanes)
        #pragma unroll
        for (int j = 0; j < 8; ++j) {
            float v = acc[j];
            int off = (row0 + j) * HDIM + m;
            y[off]    = v;  // message value
            acc1[off] = v;  // self-loop seeds the accumulator
        }
    } else {
        // tail tile only
        #pragma unroll
        for (int j = 0; j < 8; ++j) {
            int row = row0 + j;
            if (row < n) {
                float v = acc[j];
                int off = row * HDIM + m;
                y[off]    = v;
                acc1[off] = v;
            }
        }
    }
}

// ---------------- layer-1 scatter: acc1[c][f] += y[r][f], 16 lanes per edge ----------------
__global__ void k_scatter16(const int* __restrict__ row, const int* __restrict__ col,
                            const float* __restrict__ y, float* __restrict__ acc1, int E) {
    long long t = (long long)blockIdx.x * blockDim.x + threadIdx.x;
    int e = (int)(t >> 4);
    int f = (int)(t & 15);
    if (e < E) {
        int r = row[e];
        int c = col[e];
        atomicAdd(&acc1[c * HDIM + f], y[r * HDIM + f]);
    }
}

// ---------------- h = relu(b0 + dis*acc1); z = dis * (h @ W2); seed acc2 ----------------
__global__ void k_layer2(const float* __restrict__ acc1, const float* __restrict__ dis,
                         const float* __restrict__ b0, const float* __restrict__ W2,
                         float* __restrict__ z, float* __restrict__ acc2, int n) {
    int i = blockIdx.x * blockDim.x + threadIdx.x;
    if (i >= n) return;
    float dsc = dis[i];
    float s = 0.0f;
    #pragma unroll
    for (int f = 0; f < HDIM; ++f) {
        float hf = relu_f(b0[f] + dsc * acc1[i * HDIM + f]);
        s += hf * W2[f];
    }
    float zi = dsc * s;
    z[i]    = zi;
    acc2[i] = zi;  // self loop
}

// ---------------- layer-2 scatter: acc2[c] += z[r] ----------------
__global__ void k_scatter1(const int* __restrict__ row, const int* __restrict__ col,
                           const float* __restrict__ z, float* __restrict__ acc2, int E) {
    int e = blockIdx.x * blockDim.x + threadIdx.x;
    if (e < E) atomicAdd(&acc2[col[e]], z[row[e]]);
}

// ---------------- out = b2 + dis * acc2 ----------------
__global__ void k_final(const float* __restrict__ acc2, const float* __restrict__ dis,
                        const float* __restrict__ b2, float* __restrict__ out, int n) {
    int i = blockIdx.x * blockDim.x + threadIdx.x;
    if (i < n) out[i] = b2[0] + dis[i] * acc2[i];
}

extern "C" void kernel_launch(void* const* d_in, const int* in_sizes, int n_in,
                              void* d_out, int out_size, void* d_ws, size_t ws_size,
                              hipStream_t stream) {
    const float* emb_a = (const float*)d_in[0];
    const float* emb_k = (const float*)d_in[1];
    const float* W_u   = (const float*)d_in[2];
    const float* b_u   = (const float*)d_in[3];
    const float* emb_p = (const float*)d_in[4];
    const float* W_p   = (const float*)d_in[5];
    const float* b_p   = (const float*)d_in[6];
    const float* emb_c = (const float*)d_in[7];
    const float* W_c   = (const float*)d_in[8];
    const float* b_c   = (const float*)d_in[9];
    const float* W0    = (const float*)d_in[10];
    const float* b0    = (const float*)d_in[11];
    const float* W2    = (const float*)d_in[12];
    const float* b2    = (const float*)d_in[13];
    const int*   edges = (const int*)d_in[14];
    const int*   feat  = (const int*)d_in[15];

    const int E = in_sizes[14] / 2;
    const int N = in_sizes[15] / 3;
    const int* row = edges;
    const int* col = edges + E;

    float* ws   = (float*)d_ws;
    float* deg  = ws;              // N
    float* dis  = ws + (size_t)N;  // N
    float* x    = ws + 2ull * N;   // 8N
    float* y    = ws + 10ull * N;  // 16N
    float* acc1 = ws + 26ull * N;  // 16N
    float* z    = ws + 42ull * N;  // N
    float* acc2 = ws + 43ull * N;  // N   (44N floats total)

    const int T = 256;
    dim3 blk(T);

    // 1) degrees (self loop = 1 + incoming edges at col)
    k_init_deg<<<(N + T - 1) / T, blk, 0, stream>>>(deg, N);
    k_count_deg<<<(E + T - 1) / T, blk, 0, stream>>>(col, deg, E);

    // 2) typed embedding -> x, dis = rsqrt(deg)
    k_embed<<<(N + T - 1) / T, blk, 0, stream>>>(emb_a, emb_k, W_u, b_u, emb_p, W_p, b_p,
                                                 emb_c, W_c, b_c, feat, deg, dis, x, N);

    // 3) y = dis * (x @ W0) via WMMA; acc1 seeded with self-loop term
    {
        int waves  = (N + 15) / 16;
        int wpb    = T / 32;
        int blocks = (waves + wpb - 1) / wpb;
        k_xw0_wmma<<<blocks, blk, 0, stream>>>(x, dis, W0, y, acc1, N);
    }

    // 4) layer-1 edge scatter (16 lanes per edge, coalesced)
    {
        long long total = (long long)E * HDIM;
        int blocks = (int)((total + T - 1) / T);
        k_scatter16<<<blocks, blk, 0, stream>>>(row, col, y, acc1, E);
    }

    // 5) h = relu(b0 + dis*acc1); z = dis*(h @ W2); acc2 seeded
    k_layer2<<<(N + T - 1) / T, blk, 0, stream>>>(acc1, dis, b0, W2, z, acc2, N);

    // 6) layer-2 edge scatter
    k_scatter1<<<(E + T - 1) / T, blk, 0, stream>>>(row, col, z, acc2, E);

    // 7) out = b2 + dis * acc2
    k_final<<<(N + T - 1) / T, blk, 0, stream>>>(acc2, dis, b2, (float*)d_out, N);
}